// Wiener_18528488915108
// MI455X (gfx1250) — compile-verified
//
#include <hip/hip_runtime.h>

// ---------------------------------------------------------------------------
// Wiener 3x3 filter, 4096x4096 fp32, MI455X (gfx1250).
//
// Memory-bound (AI ~2.5 flop/B): floor ~5.5us at 23.3 TB/s for 64MB in +
// 64MB out (pass-2 img re-read should hit the 192MB global L2).
// CDNA5 path: Tensor Data Mover (tensor_load_to_lds) DMAs each block's
// 258x18 halo tile into LDS (iterate mode keeps LDS pitch fixed), tracked by
// TENSORcnt; compute is a rolling separable box-sum from LDS; output uses
// non-temporal stores to keep img resident in L2.
//
// Kernels are non-template and defined pass1-first so the disasm snippet
// shows the TDM descriptor build + tensor_load_to_lds.
// ---------------------------------------------------------------------------

#define WIE_W 4096
#define WIE_H 4096
#define WIE_TX 256
#define WIE_TY 16
#define WIE_ROWS (WIE_TY + 2)   // 18
#define WIE_COLS (WIE_TX + 2)   // 258
#define WIE_PITCH 260           // LDS row pitch (elements)

#if defined(__HIP_DEVICE_COMPILE__) && defined(__gfx1250__) && \
    __has_builtin(__builtin_amdgcn_tensor_load_to_lds) &&      \
    __has_builtin(__builtin_amdgcn_s_wait_tensorcnt)
#define WIE_TDM 1
#else
#define WIE_TDM 0
#endif

#if WIE_TDM
typedef unsigned int wie_u32x4 __attribute__((ext_vector_type(4)));
typedef int wie_i32x4 __attribute__((ext_vector_type(4)));
typedef int wie_i32x8 __attribute__((ext_vector_type(8)));
#endif

template <bool PASS2>
__device__ __forceinline__
void wiener_tile_body(const float* __restrict__ img,
                      float* __restrict__ out,
                      float* __restrict__ partial,
                      const float* __restrict__ noise_p,
                      float* __restrict__ tile /* LDS, WIE_ROWS*WIE_PITCH */)
{
    const int tid = threadIdx.x;
    const int bx  = blockIdx.x;
    const int by  = blockIdx.y;
    const int gx0 = bx * WIE_TX - 1;   // global col of tile col 0 (may be -1)
    const int gy0 = by * WIE_TY - 1;   // global row of tile row 0 (may be -1)

#if WIE_TDM
    {
        const int ox = (gx0 < 0) ? 1 : 0;            // left halo clamp
        const int oy = (gy0 < 0) ? 1 : 0;            // top halo clamp
        const int sx = gx0 + ox;
        const int sy = gy0 + oy;
        int h_valid = WIE_ROWS - oy;                 // rows the DMA can supply
        if (h_valid > WIE_H - sy) h_valid = WIE_H - sy;

        // Zero halo cells the TDM cannot produce (disjoint from DMA target).
        if (ox) for (int r = tid; r < WIE_ROWS; r += 256) tile[r * WIE_PITCH] = 0.0f;
        if (oy) for (int k = tid; k < WIE_COLS; k += 256) tile[k] = 0.0f;
        for (int r = oy + h_valid; r < WIE_ROWS; ++r)
            for (int k = tid; k < WIE_COLS; k += 256) tile[r * WIE_PITCH + k] = 0.0f;

        if (tid == 0) {
            // LDS byte offset = low 32 bits of flat LDS address.
            const unsigned lds_base =
                (unsigned)(unsigned long long)(const void*)&tile[0];
            const unsigned lds_addr =
                lds_base + (unsigned)((oy * WIE_PITCH + ox) * 4);
            const unsigned long long ga =
                (unsigned long long)(const void*)img +
                ((unsigned long long)(unsigned)sy * WIE_W + (unsigned)sx) * 4ull;

            const unsigned td0 = (unsigned)(WIE_W - sx);  // elems to right edge (OOB -> 0)
            const unsigned td1 = (unsigned)(WIE_H - sy);  // rows to bottom edge
            const unsigned w   = (unsigned)(WIE_COLS - ox);

            // ---- D# group 0 (128b): count=1 | lds_addr | global_addr | type=2
            wie_u32x4 g0;
            g0.x = 1u;
            g0.y = lds_addr;
            g0.z = (unsigned)ga;
            g0.w = (unsigned)(ga >> 32) | (2u << 30);

            // ---- D# group 1 (256b)
            wie_i32x8 g1;
            g1[0] = (int)((2u << 16) | (1u << 19));                 // data_size=4B, iterate_enable
            g1[1] = (int)((td0 & 0xFFFFu) << 16);                   // tensor_dim0[15:0]
            g1[2] = (int)((td0 >> 16) | ((td1 & 0xFFFFu) << 16));   // td0 hi | tensor_dim1 lo
            g1[3] = (int)((td1 >> 16) | ((w & 0xFFFFu) << 16));     // td1 hi | tile_dim0 = w
            g1[4] = 1;                                              // tile_dim1 = 1 row / iter
            g1[5] = WIE_W;                                          // tensor_dim0_stride = 4096
            g1[6] = (int)((unsigned)WIE_W << 16);                   // tensor_dim1_stride lo
            g1[7] = 0;

            // ---- D# group 2 (128b, iterate mode)
            wie_i32x4 g2;
            g2[0] = 0;                                      // tensor_dim2
            g2[1] = WIE_PITCH;                              // lds_addr_increment (elements)
            g2[2] = WIE_W;                                  // global_addr_increment (elements)
            g2[3] = (int)(((unsigned)(h_valid - 1)) << 16); // iterate_count = rows-1

            wie_i32x4 g3 = {0, 0, 0, 0};

#if defined(__clang_major__) && (__clang_major__ >= 23)
            wie_i32x8 g4 = {0, 0, 0, 0, 0, 0, 0, 0};
            __builtin_amdgcn_tensor_load_to_lds(g0, g1, g2, g3, g4, 0);
#else
            __builtin_amdgcn_tensor_load_to_lds(g0, g1, g2, g3, 0);
#endif
            __builtin_amdgcn_s_wait_tensorcnt(0);
        }
        __syncthreads();
    }
#else
    // Fallback: cooperative staging with bounds -> 0 (zero padding).
    for (int idx = tid; idx < WIE_ROWS * WIE_COLS; idx += 256) {
        const int r  = idx / WIE_COLS;
        const int k  = idx - r * WIE_COLS;
        const int gr = gy0 + r;
        const int gc = gx0 + k;
        float v = 0.0f;
        if ((unsigned)gr < (unsigned)WIE_H && (unsigned)gc < (unsigned)WIE_W)
            v = img[(long)gr * WIE_W + gc];
        tile[r * WIE_PITCH + k] = v;
    }
    __syncthreads();
#endif

    // ------------------------------------------------------------------
    // Rolling separable 3x3 box sums: thread owns one output column.
    // ------------------------------------------------------------------
    const int c  = tid;                  // window left col in tile
    const int gc = bx * WIE_TX + tid;    // global output column
    const float inv9 = 1.0f / 9.0f;

    float noise = 0.0f;
    if constexpr (PASS2) noise = noise_p[0];

    float hx0, hxx0, hx1, hxx1, m1;
    {
        float a = tile[0 * WIE_PITCH + c];
        float b = tile[0 * WIE_PITCH + c + 1];
        float d = tile[0 * WIE_PITCH + c + 2];
        hx0 = a + b + d;
        hxx0 = a * a + b * b + d * d;
        a = tile[1 * WIE_PITCH + c];
        b = tile[1 * WIE_PITCH + c + 1];
        d = tile[1 * WIE_PITCH + c + 2];
        hx1 = a + b + d;
        hxx1 = a * a + b * b + d * d;
        m1 = b;                          // center pixel of row 1
    }

    float lvar_sum = 0.0f;
    for (int y = 0; y < WIE_TY; ++y) {
        const float a = tile[(y + 2) * WIE_PITCH + c];
        const float b = tile[(y + 2) * WIE_PITCH + c + 1];
        const float d = tile[(y + 2) * WIE_PITCH + c + 2];
        const float hx2  = a + b + d;
        const float hxx2 = a * a + b * b + d * d;

        const float s1    = hx0 + hx1 + hx2;
        const float s2    = hxx0 + hxx1 + hxx2;
        const float lmean = s1 * inv9;
        const float lvar  = s2 * inv9 - lmean * lmean;

        if constexpr (PASS2) {
            const float x = m1;          // center pixel (row y+1)
            float res;
            if (lvar < noise) res = lmean;
            else              res = lmean + (1.0f - noise / lvar) * (x - lmean);
            __builtin_nontemporal_store(res, &out[(long)(by * WIE_TY + y) * WIE_W + gc]);
        } else {
            lvar_sum += lvar;
        }

        hx0 = hx1; hxx0 = hxx1;
        hx1 = hx2; hxx1 = hxx2;
        m1 = b;
    }

    if constexpr (!PASS2) {
        // Deterministic per-block reduction of sum(lvar); reuse the tile LDS.
        __syncthreads();
        tile[tid] = lvar_sum;
        __syncthreads();
        for (int s = 128; s > 0; s >>= 1) {
            if (tid < s) tile[tid] += tile[tid + s];
            __syncthreads();
        }
        if (tid == 0) partial[by * gridDim.x + bx] = tile[0];
    }
}

// ---- Pass 1: per-block sum(lvar) ------------------------------------------
__global__ __launch_bounds__(256)
void wiener_pass1_kernel(const float* __restrict__ img,
                         float* __restrict__ partial)
{
    __shared__ float tile[WIE_ROWS * WIE_PITCH];
    wiener_tile_body<false>(img, nullptr, partial, nullptr, tile);
}

// ---- Pass 2: Wiener blend --------------------------------------------------
__global__ __launch_bounds__(256)
void wiener_pass2_kernel(const float* __restrict__ img,
                         float* __restrict__ out,
                         const float* __restrict__ noise_p)
{
    __shared__ float tile[WIE_ROWS * WIE_PITCH];
    wiener_tile_body<true>(img, out, nullptr, noise_p, tile);
}

// ---- Noise = mean(lvar) ----------------------------------------------------
__global__ __launch_bounds__(256)
void wiener_noise_reduce(const float* __restrict__ partial,
                         float* __restrict__ noise_out, int n)
{
    __shared__ double red[256];
    double s = 0.0;
    for (int i = threadIdx.x; i < n; i += 256) s += (double)partial[i];
    red[threadIdx.x] = s;
    __syncthreads();
    for (int k = 128; k > 0; k >>= 1) {
        if ((int)threadIdx.x < k) red[threadIdx.x] += red[threadIdx.x + k];
        __syncthreads();
    }
    if (threadIdx.x == 0)
        noise_out[0] = (float)(red[0] * (1.0 / ((double)WIE_W * (double)WIE_H)));
}

extern "C" void kernel_launch(void* const* d_in, const int* in_sizes, int n_in,
                              void* d_out, int out_size, void* d_ws, size_t ws_size,
                              hipStream_t stream)
{
    (void)in_sizes; (void)n_in; (void)out_size; (void)ws_size;

    const float* img = (const float*)d_in[0];
    float* out = (float*)d_out;
    float* ws  = (float*)d_ws;

    float* noise   = ws;        // ws[0]: noise scalar
    float* partial = ws + 16;   // ws[16..16+4096): per-block lvar partial sums

    dim3 grid(WIE_W / WIE_TX, WIE_H / WIE_TY);  // 16 x 256 = 4096 blocks
    dim3 block(256);

    // Pass 1: local variance sums (img becomes L2-resident).
    wiener_pass1_kernel<<<grid, block, 0, stream>>>(img, partial);
    // Noise = mean(lvar).
    wiener_noise_reduce<<<1, 256, 0, stream>>>(partial, noise, (int)(grid.x * grid.y));
    // Pass 2: Wiener blend (img from L2, NT stores for output).
    wiener_pass2_kernel<<<grid, block, 0, stream>>>(img, out, noise);
}